// BiCutLoss_52312701665760
// MI455X (gfx1250) — compile-verified
//
#include <hip/hip_runtime.h>

// BiCutLoss: single-pass, HBM-bound streaming reduction.
// output: [4096, 4096, 2] f32 (interleaved ch0/ch1), labels: [4096,4096] i32.
// One 256-thread block per row; contributions held in registers across the
// cut-index barrier so the 192 MiB input is read exactly once.

typedef float v4f __attribute__((ext_vector_type(4)));
typedef int   v2i __attribute__((ext_vector_type(2)));

constexpr int B_DIM   = 4096;
constexpr int L_DIM   = 4096;
constexpr int THREADS = 256;                    // 8 wave32s
constexpr int PAIRS   = L_DIM / (2 * THREADS);  // 8 b128 loads / thread (2 elems each)
constexpr float NEG_REWARD = 0.065f;            // ALPHA * 0.1

__device__ __forceinline__ int wave_max_i(int v) {
#pragma unroll
  for (int m = 16; m >= 1; m >>= 1) v = max(v, __shfl_xor(v, m, 32));
  return v;
}

__device__ __forceinline__ float wave_sum_f(float v) {
#pragma unroll
  for (int m = 16; m >= 1; m >>= 1) v += __shfl_xor(v, m, 32);
  return v;
}

// pos_reward(j) = -3.6 / log2(j + 2), branch-free: v_log_f32 + v_rcp_f32 + v_mul.
__device__ __forceinline__ float pos_reward(float j_plus) {
  return -3.6f * __builtin_amdgcn_rcpf(__log2f(j_plus));
}

__global__ __launch_bounds__(THREADS) void bicut_row_kernel(
    const float* __restrict__ out, const int* __restrict__ labels,
    float* __restrict__ row_part) {
  const int row = blockIdx.x;
  const int tid = threadIdx.x;

  // One v4f = two consecutive elements: {out0[j], out1[j], out0[j+1], out1[j+1]}
  const v4f* rowp = (const v4f*)(out + (size_t)row * (size_t)L_DIM * 2u);
  const v2i* labp = (const v2i*)(labels + (size_t)row * (size_t)L_DIM);

  float contrib[2 * PAIRS];
  int lastZero = -1;  // largest j in this thread's set with temp[j]==0

#pragma unroll
  for (int k = 0; k < PAIRS; ++k) {
    const int j0 = 2 * tid + k * (2 * THREADS);   // strictly increasing in k
    const v4f v  = __builtin_nontemporal_load(rowp + (j0 >> 1));  // b128, NT
    const v2i lb = __builtin_nontemporal_load(labp + (j0 >> 1));  // b64,  NT

    // r1[j] = pos_reward(j) if label==1 else 0.065 — unconditional + v_cndmask,
    // no execz branches.
    const float pa = pos_reward((float)j0 + 2.0f);
    const float pb = pos_reward((float)j0 + 3.0f);
    const float ra = (lb.x == 1) ? pa : NEG_REWARD;
    const float rb = (lb.y == 1) ? pb : NEG_REWARD;
    contrib[2 * k]     = v.y * ra;
    contrib[2 * k + 1] = v.w * rb;

    // temp[j]==0  <=>  out1 <= out0 (argmax tie-breaks to index 0).
    // j grows with k, and j0+1 > j0, so plain overwrite keeps the max.
    if (v.y <= v.x) lastZero = j0;
    if (v.w <= v.z) lastZero = j0 + 1;
  }

  __shared__ int   s_max[THREADS / 32];
  __shared__ float s_sum[THREADS / 32];
  const int wave = tid >> 5, lane = tid & 31;

  // Block-wide max of lastZero -> cut. cut = L if the row has no zero.
  const int wm = wave_max_i(lastZero);
  if (lane == 0) s_max[wave] = wm;
  __syncthreads();
  int bm = s_max[0];
#pragma unroll
  for (int i = 1; i < THREADS / 32; ++i) bm = max(bm, s_max[i]);
  const int cut = (bm < 0) ? L_DIM : bm;  // mask[j] = (j < cut)

  // Apply mask to register-resident contributions; fixed-order sums.
  float s = 0.0f;
#pragma unroll
  for (int k = 0; k < PAIRS; ++k) {
    const int j0 = 2 * tid + k * (2 * THREADS);
    if (j0 < cut)     s += contrib[2 * k];
    if (j0 + 1 < cut) s += contrib[2 * k + 1];
  }
  const float ws = wave_sum_f(s);
  if (lane == 0) s_sum[wave] = ws;
  __syncthreads();
  if (tid == 0) {
    float t = 0.0f;
#pragma unroll
    for (int i = 0; i < THREADS / 32; ++i) t += s_sum[i];
    row_part[row] = t;
  }
}

__global__ __launch_bounds__(256) void bicut_final_kernel(
    const float* __restrict__ row_part, float* __restrict__ outv) {
  const int tid = threadIdx.x;
  float s = 0.0f;
#pragma unroll
  for (int k = 0; k < B_DIM / 256; ++k) s += row_part[tid + k * 256];

  const int wave = tid >> 5, lane = tid & 31;
  const float ws = wave_sum_f(s);
  __shared__ float sm[8];
  if (lane == 0) sm[wave] = ws;
  __syncthreads();
  if (tid == 0) {
    float t = 0.0f;
#pragma unroll
    for (int i = 0; i < 8; ++i) t += sm[i];
    outv[0] = t * (1.0f / (float)B_DIM);  // / Bn
  }
}

extern "C" void kernel_launch(void* const* d_in, const int* in_sizes, int n_in,
                              void* d_out, int out_size, void* d_ws, size_t ws_size,
                              hipStream_t stream) {
  (void)in_sizes; (void)n_in; (void)out_size; (void)ws_size;
  const float* output = (const float*)d_in[0];  // [B, L, 2] f32
  const int*   labels = (const int*)d_in[1];    // [B, L]    i32
  float* row_part = (float*)d_ws;               // 4096 floats = 16 KiB scratch

  bicut_row_kernel<<<B_DIM, THREADS, 0, stream>>>(output, labels, row_part);
  bicut_final_kernel<<<1, 256, 0, stream>>>(row_part, (float*)d_out);
}